// VectorPoolLocalInterpolateModule_43645457662573
// MI455X (gfx1250) — compile-verified
//
#include <hip/hip_runtime.h>
#include <math.h>

typedef float v2f __attribute__((ext_vector_type(2)));
typedef float v8f __attribute__((ext_vector_type(8)));

__device__ __forceinline__ void top3_insert(float& d0, float& d1, float& d2,
                                            int& i0, int& i1, int& i2,
                                            float d, int i) {
    if (d < d2) {
        if (d < d0)      { d2 = d1; i2 = i1; d1 = d0; i1 = i0; d0 = d; i0 = i; }
        else if (d < d1) { d2 = d1; i2 = i1; d1 = d;  i1 = i; }
        else             { d2 = d;  i2 = i; }
    }
}

// ---------------------------------------------------------------------------
// Kernel 1: cube-neighborhood 3-NN + inverse-distance interpolation.
// 4 lanes per (query m, grid-center g): each lane scans 1/4 of the batch's
// support range; support points are staged through LDS in 256-point tiles
// (double-buffered via registers), then the four sorted top-3 lists are
// merged in-register with wave32 __shfl_xor butterflies (mask 1, 2).
// Output row = [interp(C=32) | local_xyz(9) | pad -> Kp=48] into d_ws.
// ---------------------------------------------------------------------------
#define TILE 256

__global__ __launch_bounds__(256) void nn_interp_kernel(
    const float* __restrict__ sx,       // support_xyz [N,3]
    const float* __restrict__ sf,       // support_features [N,C]
    const int*   __restrict__ cnt_s,    // batch_num_xyzs [Bn]
    const float* __restrict__ qx,       // new_xyz [M,3]
    const float* __restrict__ centers,  // grid centers [M,G,3]
    const int*   __restrict__ cnt_q,    // batch_num_new_xyzs [Bn]
    float* __restrict__ X,              // out: padded features [M*G, Kp]
    int M, int G, int C, int Bn, int Kp)
{
    __shared__ float tile[TILE * 3];

    int tid   = threadIdx.x;
    int t     = blockIdx.x * blockDim.x + tid;
    int total = M * G * 4;
    if (t >= total) t = total - 1;      // clamp: duplicate work is benign
    int mg    = t >> 2;
    int chunk = t & 3;
    int m     = mg / G;

    // ---- block-uniform support window (batches are contiguous in m) ----
    int mg_lo = (blockIdx.x * (int)blockDim.x) >> 2;
    int mg_hi = ((blockIdx.x + 1) * (int)blockDim.x - 1) >> 2;
    if (mg_hi >= M * G) mg_hi = M * G - 1;
    int m_lo = mg_lo / G, m_hi = mg_hi / G;

    int b_lo = Bn - 1, b_hi = Bn - 1;
    {
        int cum = 0;
        for (int b = 0; b < Bn; ++b) {
            int nc = cum + cnt_q[b];
            if (m_lo >= cum && m_lo < nc && b_lo == Bn - 1) b_lo = b;
            if (m_hi >= cum && m_hi < nc && b_hi == Bn - 1) b_hi = b;
            cum = nc;
        }
        // re-derive cleanly (small Bn, cheap):
        cum = 0; b_lo = Bn - 1;
        for (int b = 0; b < Bn; ++b) { cum += cnt_q[b]; if (m_lo < cum) { b_lo = b; break; } }
        cum = 0; b_hi = Bn - 1;
        for (int b = 0; b < Bn; ++b) { cum += cnt_q[b]; if (m_hi < cum) { b_hi = b; break; } }
    }
    int sstart_blk = 0;
    for (int b = 0; b < b_lo; ++b) sstart_blk += cnt_s[b];
    int send_blk = sstart_blk;
    for (int b = b_lo; b <= b_hi; ++b) send_blk += cnt_s[b];

    // ---- per-thread batch window and chunk sub-range ----
    int batch = Bn - 1;
    {
        int cum = 0;
        for (int b = 0; b < Bn; ++b) { cum += cnt_q[b]; if (m < cum) { batch = b; break; } }
    }
    int sstart = 0;
    for (int b = 0; b < batch; ++b) sstart += cnt_s[b];
    int send = sstart + cnt_s[batch];
    int clen = (send - sstart + 3) >> 2;
    int c0 = sstart + chunk * clen;
    int c1 = min(c0 + clen, send);

    float qxx = qx[3 * m], qyy = qx[3 * m + 1], qzz = qx[3 * m + 2];
    const float* cp = centers + (size_t)mg * 3;
    float cx = cp[0], cy = cp[1], cz = cp[2];

    const float RAD = 4.0f;                  // MAX_DIST * DIST_MULT
    const float BIG = 3.402823466e38f;       // sentinel for unfilled slot
    float d0 = BIG, d1 = BIG, d2 = BIG;
    int   i0 = 0,  i1 = 0,  i2 = 0;

    // ---- tiled scan: global -> regs -> LDS, process from LDS ----
    float rx = 0.0f, ry = 0.0f, rz = 0.0f;
    {
        int gi = sstart_blk + tid;
        if (gi < send_blk) { rx = sx[3 * gi]; ry = sx[3 * gi + 1]; rz = sx[3 * gi + 2]; }
    }
    for (int base = sstart_blk; base < send_blk; base += TILE) {
        __syncthreads();                       // previous tile fully consumed
        tile[tid * 3 + 0] = rx;
        tile[tid * 3 + 1] = ry;
        tile[tid * 3 + 2] = rz;
        __syncthreads();                       // tile visible to all waves
        // prefetch next tile into registers (hides global latency)
        int gn = base + TILE + tid;
        rx = ry = rz = 0.0f;
        if (gn < send_blk) { rx = sx[3 * gn]; ry = sx[3 * gn + 1]; rz = sx[3 * gn + 2]; }
        // process intersection of this tile with my chunk range
        int lo = max(c0, base), hi = min(c1, base + TILE);
        for (int i = lo; i < hi; ++i) {
            int j = (i - base) * 3;
            float px = tile[j], py = tile[j + 1], pz = tile[j + 2];
            if (fabsf(px - qxx) < RAD && fabsf(py - qyy) < RAD && fabsf(pz - qzz) < RAD) {
                float dx = cx - px, dy = cy - py, dz = cz - pz;
                top3_insert(d0, d1, d2, i0, i1, i2, dx * dx + dy * dy + dz * dz, i);
            }
        }
    }

    // ---- merge the 4 chunk lists with wave32 xor-butterflies ----
    #pragma unroll
    for (int mask = 1; mask <= 2; mask <<= 1) {
        float e0 = __shfl_xor(d0, mask, 32);
        float e1 = __shfl_xor(d1, mask, 32);
        float e2 = __shfl_xor(d2, mask, 32);
        int   j0 = __shfl_xor(i0, mask, 32);
        int   j1 = __shfl_xor(i1, mask, 32);
        int   j2 = __shfl_xor(i2, mask, 32);
        top3_insert(d0, d1, d2, i0, i1, i2, e0, j0);
        top3_insert(d0, d1, d2, i0, i1, i2, e1, j1);
        top3_insert(d0, d1, d2, i0, i1, i2, e2, j2);
    }

    if (chunk != 0) return;                    // one writer per (m,g)

    float* row = X + (size_t)mg * Kp;
    bool found = (d0 < BIG);                   // any candidate across all chunks
    if (!found) {                              // empty query -> zero row (ref)
        for (int j = 0; j < Kp; ++j) row[j] = 0.0f;
        return;
    }

    float s0 = (d0 >= BIG) ? 1e8f : sqrtf(fmaxf(d0, 0.0f));
    float s1 = (d1 >= BIG) ? 1e8f : sqrtf(fmaxf(d1, 0.0f));
    float s2 = (d2 >= BIG) ? 1e8f : sqrtf(fmaxf(d2, 0.0f));
    float r0 = 1.0f / (s0 + 1e-8f), r1 = 1.0f / (s1 + 1e-8f), r2 = 1.0f / (s2 + 1e-8f);
    float inv = 1.0f / fmaxf(r0 + r1 + r2, 1e-8f);
    float w0 = r0 * inv, w1 = r1 * inv, w2 = r2 * inv;

    const float* f0 = sf + (size_t)i0 * C;
    const float* f1 = sf + (size_t)i1 * C;
    const float* f2 = sf + (size_t)i2 * C;
    for (int c = 0; c < C; ++c)
        row[c] = w0 * f0[c] + w1 * f1[c] + w2 * f2[c];

    int idxs[3] = { i0, i1, i2 };
    #pragma unroll
    for (int k = 0; k < 3; ++k) {
        const float* p = sx + (size_t)idxs[k] * 3;
        row[C + 3 * k + 0] = cx - p[0];
        row[C + 3 * k + 1] = cy - p[1];
        row[C + 3 * k + 2] = cz - p[2];
    }
    for (int j = C + 9; j < Kp; ++j) row[j] = 0.0f;
}

// ---------------------------------------------------------------------------
// Kernel 2: [R x 48] @ [48 x 128] GEMM via V_WMMA_F32_16X16X4_F32, with BN
// scale folded into B (staged in LDS) and BN shift + ReLU fused in epilogue.
// 8 waves/block; each wave owns one 16-row M-tile across all 8 N-tiles.
// ---------------------------------------------------------------------------
#define KP   48
#define COUT 128

__global__ __launch_bounds__(256) void gemm_bn_relu_kernel(
    const float* __restrict__ X,      // [R, KP]
    const float* __restrict__ W,      // conv_w [COUT, Kin]
    const float* __restrict__ gamma,
    const float* __restrict__ beta,
    const float* __restrict__ mean,
    const float* __restrict__ var,
    float* __restrict__ out,          // [R, COUT]
    int R, int Kin)
{
    __shared__ float Blds[KP * COUT];   // B with BN scale folded in
    __shared__ float Shift[COUT];       // beta - scale*mean

    int tid = threadIdx.x;
    for (int idx = tid; idx < KP * COUT; idx += 256) {
        int k = idx / COUT, n = idx % COUT;
        float sc = gamma[n] * rsqrtf(var[n] + 1e-5f);
        Blds[idx] = (k < Kin) ? W[(size_t)n * Kin + k] * sc : 0.0f;
    }
    for (int n = tid; n < COUT; n += 256) {
        float sc = gamma[n] * rsqrtf(var[n] + 1e-5f);
        Shift[n] = beta[n] - sc * mean[n];
    }
    __syncthreads();

    int wave  = tid >> 5;
    int lane  = tid & 31;
    int lrow  = lane & 15;
    int khalf = lane >> 4;                 // 0: K pair {0,1}, 1: K pair {2,3}
    int m0    = (blockIdx.x * 8 + wave) * 16;

    if (m0 + 16 <= R) {
        v8f acc[8] = {};
        const float* xrow = X + (size_t)(m0 + lrow) * KP + khalf * 2;

        for (int k0 = 0; k0 < KP; k0 += 4) {
            v2f a;
            a.x = xrow[k0];
            a.y = xrow[k0 + 1];
            int kb = k0 + khalf * 2;
            #pragma unroll
            for (int nt = 0; nt < 8; ++nt) {
                v2f b;
                b.x = Blds[kb * COUT + nt * 16 + lrow];
                b.y = Blds[(kb + 1) * COUT + nt * 16 + lrow];
                acc[nt] = __builtin_amdgcn_wmma_f32_16x16x4_f32(
                    /*neg_a=*/false, a, /*neg_b=*/false, b,
                    /*c_mod=*/(short)0, acc[nt],
                    /*reuse_a=*/false, /*reuse_b=*/false);
            }
        }

        #pragma unroll
        for (int nt = 0; nt < 8; ++nt) {
            int col = nt * 16 + lrow;
            float sh = Shift[col];
            #pragma unroll
            for (int v = 0; v < 8; ++v) {
                int rrow = m0 + v + khalf * 8;
                float y = acc[nt][v] + sh;
                out[(size_t)rrow * COUT + col] = fmaxf(y, 0.0f);
            }
        }
    }
}

extern "C" void kernel_launch(void* const* d_in, const int* in_sizes, int n_in,
                              void* d_out, int out_size, void* d_ws, size_t ws_size,
                              hipStream_t stream) {
    const float* support_xyz      = (const float*)d_in[0];
    const float* support_features = (const float*)d_in[1];
    const int*   batch_num_xyzs   = (const int*)d_in[2];
    const float* new_xyz          = (const float*)d_in[3];
    const float* grid_centers     = (const float*)d_in[4];
    const int*   batch_num_new    = (const int*)d_in[5];
    const float* conv_w           = (const float*)d_in[6];
    const float* bn_gamma         = (const float*)d_in[7];
    const float* bn_beta          = (const float*)d_in[8];
    const float* bn_mean          = (const float*)d_in[9];
    const float* bn_var           = (const float*)d_in[10];
    float* out = (float*)d_out;

    int N    = in_sizes[0] / 3;                 // 8192 support points
    int Bn   = in_sizes[2];                     // 2 batches
    int M    = in_sizes[3] / 3;                 // 2048 query points
    int G    = in_sizes[4] / in_sizes[3];       // 8 grid centers
    int C    = in_sizes[1] / N;                 // 32 feature channels
    int Kin  = C + 9;                           // 41
    (void)N; (void)ws_size; (void)n_in; (void)out_size;

    float* Xpad = (float*)d_ws;                 // [M*G, KP] padded features
    int total = M * G;                          // 16384 rows

    int threads1 = total * 4;                   // 4 chunk-lanes per (m,g)
    nn_interp_kernel<<<(threads1 + 255) / 256, 256, 0, stream>>>(
        support_xyz, support_features, batch_num_xyzs, new_xyz,
        grid_centers, batch_num_new, Xpad, M, G, C, Bn, KP);

    int mtiles = (total + 15) / 16;             // 1024
    int blocks = (mtiles + 7) / 8;              // 128 (8 waves per block)
    gemm_bn_relu_kernel<<<blocks, 256, 0, stream>>>(
        Xpad, conv_w, bn_gamma, bn_beta, bn_mean, bn_var, out, total, Kin);
}